// PointLoss_test_1408749273446
// MI455X (gfx1250) — compile-verified
//
#include <hip/hip_runtime.h>

typedef __attribute__((ext_vector_type(2))) float v2f;
typedef __attribute__((ext_vector_type(4))) float v4f;
typedef __attribute__((ext_vector_type(8))) float v8f;

#define NPTS   4096
#define NTILES (NPTS / 16)      // 256 column tiles
#define NBATCH 8
#define TOTAL_PTS (NBATCH * NPTS)   // 32768 points per array
#define BLOCK_THREADS 256           // 8 waves per block (wave32)
#define WAVES_PER_BLOCK (BLOCK_THREADS / 32)

__global__ void chamfer_init(float* ws) {
    ws[0] = 0.0f;
    ws[1] = 0.0f;
}

// Pack (x,y,z) -> (x,y,z,||p||^2) for both arrays (pre-computed once so the
// hot loop does zero norm/selection VALU work).
__global__ __launch_bounds__(256)
void chamfer_pack(const float* __restrict__ a1, const float* __restrict__ a2,
                  v4f* __restrict__ p1, v4f* __restrict__ p2) {
    const int idx   = blockIdx.x * blockDim.x + threadIdx.x;   // 0 .. 2*TOTAL_PTS-1
    const int which = idx / TOTAL_PTS;
    const int p     = idx - which * TOTAL_PTS;
    const float* __restrict__ src = which ? a2 : a1;
    v4f* __restrict__ dst         = which ? p2 : p1;
    const float x = src[p * 3 + 0];
    const float y = src[p * 3 + 1];
    const float z = src[p * 3 + 2];
    v4f o;
    o.x = x; o.y = y; o.z = z; o.w = x * x + y * y + z * z;
    dst[p] = o;
}

// Shared epilogue: rmin[r] holds per-lane running mins; C layout is
// VGPR r, lanes 0-15 -> row r ; lanes 16-31 -> row r+8. Cross-lane min
// within each 16-lane half, add sum of ||p_i||^2, one atomicAdd per wave.
__device__ __forceinline__
void chamfer_epilogue(v8f rmin, float n1, int lane, float* acc) {
    float rowsum = 0.0f;
    #pragma unroll
    for (int r = 0; r < 8; ++r) {
        float v = rmin[r];
        #pragma unroll
        for (int m = 1; m <= 8; m <<= 1)
            v = fminf(v, __shfl_xor(v, m, 32));
        rowsum += v;   // lanes 0-15: rows 0..7 ; lanes 16-31: rows 8..15
    }
    float nsum = n1;   // lanes 0-15 cover all 16 rows' norms
    #pragma unroll
    for (int m = 1; m <= 8; m <<= 1)
        nsum += __shfl_xor(nsum, m, 32);

    const float total = __shfl(rowsum, 0, 32) + __shfl(rowsum, 16, 32)
                      + __shfl(nsum, 0, 32);
    if (lane == 0)
        atomicAdd(acc, total);
}

// Fast path: inputs pre-packed as float4 (x,y,z,norm). One wave owns a
// 16-row tile of P, sweeps all 256 column tiles of Q. Per tile: one b64
// load + one V_WMMA_F32_16X16X4_F32 + 8 v_min (fused to min3 by unroll).
__global__ __launch_bounds__(BLOCK_THREADS)
void chamfer_wmma_packed(const v4f* __restrict__ p1, const v4f* __restrict__ p2,
                         float* __restrict__ accum) {
    const int dir = blockIdx.y;   // 0: rows=arr2,cols=arr1 ; 1: rows=arr1,cols=arr2
    const float* __restrict__ P = (const float*)(dir ? p1 : p2);
    const float* __restrict__ Q = (const float*)(dir ? p2 : p1);
    float* acc = accum + dir;

    const int wave  = blockIdx.x * WAVES_PER_BLOCK + (threadIdx.x >> 5);
    const int b     = wave / NTILES;
    const int itile = wave % NTILES;
    const int lane  = threadIdx.x & 31;
    const int l16   = lane & 15;
    const bool upper = (lane & 16) != 0;

    const float* __restrict__ Pb = P + (size_t)b * NPTS * 4;
    const float* __restrict__ Qb = Q + (size_t)b * NPTS * 4;

    // A tile: row i K-vector = (-2x, -2y, -2z, 1); lanes 0-15 hold (K0,K1),
    // lanes 16-31 hold (K2,K3).
    const int i = itile * 16 + l16;
    const v4f p = *(const v4f*)(Pb + (size_t)i * 4);
    const float n1 = p.w;
    v2f a;
    a.x = upper ? (-2.0f * p.z) : (-2.0f * p.x);
    a.y = upper ? 1.0f          : (-2.0f * p.y);

    const float INF = __builtin_inff();
    v8f rmin = { INF, INF, INF, INF, INF, INF, INF, INF };

    // Per-lane B pointer: lower lanes read (x,y)=(K0,K1), upper read (z,n2)=(K2,K3).
    const float* __restrict__ qlane = Qb + (size_t)l16 * 4 + (upper ? 2 : 0);

    #pragma unroll 4
    for (int jt = 0; jt < NTILES; ++jt) {
        const v2f bb = *(const v2f*)(qlane + (size_t)jt * 64);   // 16 pts * 4 floats
        v8f c = { 0.f, 0.f, 0.f, 0.f, 0.f, 0.f, 0.f, 0.f };
        c = __builtin_amdgcn_wmma_f32_16x16x4_f32(
                false, a, false, bb, (short)0, c, false, false);
        #pragma unroll
        for (int r = 0; r < 8; ++r)
            rmin[r] = fminf(rmin[r], c[r]);
    }

    chamfer_epilogue(rmin, n1, lane, acc);
}

// Fallback (no workspace for packing): compute norms/selects in-loop.
__global__ __launch_bounds__(BLOCK_THREADS)
void chamfer_wmma_raw(const float* __restrict__ a1, const float* __restrict__ a2,
                      float* __restrict__ accum) {
    const int dir = blockIdx.y;
    const float* __restrict__ P = dir ? a1 : a2;
    const float* __restrict__ Q = dir ? a2 : a1;
    float* acc = accum + dir;

    const int wave  = blockIdx.x * WAVES_PER_BLOCK + (threadIdx.x >> 5);
    const int b     = wave / NTILES;
    const int itile = wave % NTILES;
    const int lane  = threadIdx.x & 31;
    const int l16   = lane & 15;
    const bool upper = (lane & 16) != 0;

    const float* __restrict__ Pb = P + (size_t)b * NPTS * 3;
    const float* __restrict__ Qb = Q + (size_t)b * NPTS * 3;

    const int i = itile * 16 + l16;
    const float x1 = Pb[i * 3 + 0];
    const float y1 = Pb[i * 3 + 1];
    const float z1 = Pb[i * 3 + 2];
    const float n1 = x1 * x1 + y1 * y1 + z1 * z1;
    v2f a;
    a.x = upper ? (-2.0f * z1) : (-2.0f * x1);
    a.y = upper ? 1.0f         : (-2.0f * y1);

    const float INF = __builtin_inff();
    v8f rmin = { INF, INF, INF, INF, INF, INF, INF, INF };

    #pragma unroll 2
    for (int jt = 0; jt < NTILES; ++jt) {
        const int j = jt * 16 + l16;
        const float x2 = Qb[j * 3 + 0];
        const float y2 = Qb[j * 3 + 1];
        const float z2 = Qb[j * 3 + 2];
        const float n2 = x2 * x2 + y2 * y2 + z2 * z2;
        v2f bb;
        bb.x = upper ? z2 : x2;
        bb.y = upper ? n2 : y2;
        v8f c = { 0.f, 0.f, 0.f, 0.f, 0.f, 0.f, 0.f, 0.f };
        c = __builtin_amdgcn_wmma_f32_16x16x4_f32(
                false, a, false, bb, (short)0, c, false, false);
        #pragma unroll
        for (int r = 0; r < 8; ++r)
            rmin[r] = fminf(rmin[r], c[r]);
    }

    chamfer_epilogue(rmin, n1, lane, acc);
}

__global__ void chamfer_finalize(const float* __restrict__ ws, float* __restrict__ out) {
    const float inv = 1.0f / ((float)NBATCH * (float)NPTS);
    const float d1 = ws[0] * inv;   // mean over (b,j) of min_i
    const float d2 = ws[1] * inv;   // mean over (b,i) of min_j
    out[0] = d1 + d2;
    out[1] = d1;
    out[2] = d2;
}

extern "C" void kernel_launch(void* const* d_in, const int* in_sizes, int n_in,
                              void* d_out, int out_size, void* d_ws, size_t ws_size,
                              hipStream_t stream) {
    const float* a1 = (const float*)d_in[0];   // array1 [8,4096,3] f32
    const float* a2 = (const float*)d_in[1];   // array2 [8,4096,3] f32
    float* ws  = (float*)d_ws;                 // ws[0]=dist1 sum, ws[1]=dist2 sum
    float* out = (float*)d_out;

    chamfer_init<<<1, 1, 0, stream>>>(ws);

    const size_t packed_bytes = (size_t)TOTAL_PTS * sizeof(v4f);   // 512 KB each
    const size_t need = 16 + 2 * packed_bytes;
    dim3 grid(NBATCH * NTILES / WAVES_PER_BLOCK, 2);               // (256, 2)

    if (ws_size >= need) {
        v4f* p1 = (v4f*)((char*)d_ws + 16);
        v4f* p2 = p1 + TOTAL_PTS;
        chamfer_pack<<<(2 * TOTAL_PTS) / 256, 256, 0, stream>>>(a1, a2, p1, p2);
        chamfer_wmma_packed<<<grid, BLOCK_THREADS, 0, stream>>>(p1, p2, ws);
    } else {
        chamfer_wmma_raw<<<grid, BLOCK_THREADS, 0, stream>>>(a1, a2, ws);
    }

    chamfer_finalize<<<1, 1, 0, stream>>>(ws, out);
}